// AdversarialDecomposer_74758200754874
// MI455X (gfx1250) — compile-verified
//
#include <hip/hip_runtime.h>
#include <math.h>

#define EMBED   300
#define HIDDEN  300
#define BATCHSZ 32768
#define KNEG    10
#define PAD     304      // 300 padded to multiple of 16
#define NTILES  19       // 304 / 16 N-tiles
#define ROWS    32       // batch rows per block (2 M-tiles)
#define MTILES  2
#define WAVES   8
#define TPB     (WAVES * 32)
#define NBLOCKS (BATCHSZ / ROWS)
#define EPSV    1e-5f

typedef __attribute__((ext_vector_type(2))) float v2f;
typedef __attribute__((ext_vector_type(8))) float v8f;
typedef __attribute__((ext_vector_type(4))) int   v4i;

#if __has_builtin(__builtin_amdgcn_global_load_async_to_lds_b128) && \
    __has_builtin(__builtin_amdgcn_s_wait_asynccnt)
#define USE_ASYNC_LDS 1
// Builtin signature (from hipcc diagnostic): arg0 = v4i addrspace(1)* (global src),
// arg1 = v4i addrspace(3)* (LDS dst), arg2/arg3 = imm offset / cpol.
typedef __attribute__((address_space(1))) v4i* g_v4i;
typedef __attribute__((address_space(3))) v4i* l_v4i;
#else
#define USE_ASYNC_LDS 0
#endif

__device__ __forceinline__ float waveRedSum(float v) {
#pragma unroll
  for (int m = 16; m >= 1; m >>= 1) v += __shfl_xor(v, m, 32);
  return v;
}

// D = A(LDS ROWSxK) * W^T + bias, via V_WMMA_F32_16X16X4_F32.
// A frag (32-bit 16x4): lanes 0-15 -> M=lane, K={k0,k0+1}; lanes 16-31 -> M=lane-16, K={k0+2,k0+3}.
// B frag mirrors with N for M. C/D: VGPR i: lanes 0-15 -> M=i, N=lane; lanes 16-31 -> M=i+8.
// Pad columns (n=300..303) read clamped row 299 unconditionally; their D values land in
// the LDS pad region and are never consumed, so no predication is needed.
__device__ __forceinline__ void wmmaGemm(
    const float* __restrict__ As,    // LDS [ROWS][PAD]
    const float* __restrict__ W,     // [300][300] row-major (out-dim x in-dim)
    const float* __restrict__ bias,  // [300]
    float* __restrict__ Ds,          // LDS [ROWS][PAD]
    int wave, int lane) {
  const int lh = lane & 15;
  const int kq = (lane >> 4) << 1;  // 0 or 2
  const int mb = (lane >> 4) << 3;  // 0 or 8 (D-matrix row base)
  for (int wt = wave; wt < MTILES * NTILES; wt += WAVES) {
    const int mt = (wt >= NTILES) ? 1 : 0;
    const int nt = wt - mt * NTILES;
    const int col = nt * 16 + lh;
    const int colc = (col < 300) ? col : 299;         // clamped, branch-free
    const float* __restrict__ wrow = W + colc * 300;
    const float* __restrict__ arow = As + (mt * 16 + lh) * PAD;
    v8f acc0 = {0.f, 0.f, 0.f, 0.f, 0.f, 0.f, 0.f, 0.f};
    v8f acc1 = {0.f, 0.f, 0.f, 0.f, 0.f, 0.f, 0.f, 0.f};
    int k0 = 0;
#pragma unroll 2
    for (; k0 + 8 <= 300; k0 += 8) {          // dual accumulators, 4 WMMA / unrolled body
      float2 av0 = *(const float2*)(arow + k0 + kq);
      float2 wv0 = *(const float2*)(wrow + k0 + kq);
      float2 av1 = *(const float2*)(arow + k0 + 4 + kq);
      float2 wv1 = *(const float2*)(wrow + k0 + 4 + kq);
      v2f a0, b0, a1, b1;
      a0.x = av0.x; a0.y = av0.y; b0.x = wv0.x; b0.y = wv0.y;
      a1.x = av1.x; a1.y = av1.y; b1.x = wv1.x; b1.y = wv1.y;
      acc0 = __builtin_amdgcn_wmma_f32_16x16x4_f32(
          false, a0, false, b0, (short)0, acc0, false, false);
      acc1 = __builtin_amdgcn_wmma_f32_16x16x4_f32(
          false, a1, false, b1, (short)0, acc1, false, false);
    }
    for (; k0 < 300; k0 += 4) {               // 1 leftover step (300 mod 8 == 4)
      float2 av = *(const float2*)(arow + k0 + kq);
      float2 wv = *(const float2*)(wrow + k0 + kq);
      v2f a, b;
      a.x = av.x; a.y = av.y; b.x = wv.x; b.y = wv.y;
      acc0 = __builtin_amdgcn_wmma_f32_16x16x4_f32(
          false, a, false, b, (short)0, acc0, false, false);
    }
    const float bv = bias[colc];
#pragma unroll
    for (int i = 0; i < 8; ++i)
      Ds[(mt * 16 + mb + i) * PAD + col] = acc0[i] + acc1[i] + bv;
  }
}

__global__ __launch_bounds__(TPB) void fused_fwd(
    const int* __restrict__ center_ids,
    const int* __restrict__ ctx_ids,
    const int* __restrict__ neg_ids,
    const int* __restrict__ labels,
    const float* __restrict__ center_emb,
    const float* __restrict__ context_emb,
    const float* __restrict__ W_enc, const float* __restrict__ b_enc,
    const float* __restrict__ W_dec, const float* __restrict__ b_dec,
    const float* __restrict__ W_cls, const float* __restrict__ b_cls,
    float* __restrict__ part) {
  __shared__ float sEmb[ROWS][PAD];
  __shared__ float sEnc[ROWS][PAD];
  __shared__ float sCV[ROWS][PAD];
  __shared__ float sD[WAVES], sC[WAVES];

  const int tid = threadIdx.x;
  const int wave = tid >> 5;
  const int lane = tid & 31;
  const int row0 = blockIdx.x * ROWS;

  // Phase 0: gather ROWS center-embedding rows into LDS (rows are 75 x 16B).
  // Prefer the gfx1250 async-to-LDS path (ASYNCcnt-tracked, no VGPR bounce).
  for (int c = tid; c < ROWS * 75; c += TPB) {
    int r = c / 75, c4 = c - r * 75;
    int wid = center_ids[row0 + r];
    const float* src = center_emb + (size_t)wid * EMBED + c4 * 4;
    float* dst = &sEmb[r][c4 * 4];
#if USE_ASYNC_LDS
    float* snc = const_cast<float*>(src);
    __builtin_amdgcn_global_load_async_to_lds_b128(
        (g_v4i)(v4i*)snc, (l_v4i)(v4i*)dst, 0, 0);
#else
    *(float4*)dst = *(const float4*)src;
#endif
  }

  // Prefetch the 352 context rows needed in phase 3 (dominant HBM traffic),
  // overlapping the pull with the two WMMA GEMM phases.
  for (int pc = tid; pc < ROWS * (KNEG + 1) * 10; pc += TPB) {
    int d = pc / 10, chunk = pc - d * 10;      // 10 x 128B covers a 1200B row
    int r = d / (KNEG + 1);
    int j = d - r * (KNEG + 1);
    int gid = (j == 0) ? ctx_ids[row0 + r]
                       : neg_ids[(row0 + r) * KNEG + (j - 1)];
    __builtin_prefetch(context_emb + (size_t)gid * EMBED + chunk * 32, 0, 3);
  }
#if USE_ASYNC_LDS
  __builtin_amdgcn_s_wait_asynccnt(0);
#endif
  __syncthreads();

  // Phase 1: enc = emb @ W_enc^T + b_enc  (fp32 WMMA)
  wmmaGemm(&sEmb[0][0], W_enc, b_enc, &sEnc[0][0], wave, lane);
  __syncthreads();
  // Phase 2: center_v = enc @ W_dec^T + b_dec
  wmmaGemm(&sEnc[0][0], W_dec, b_dec, &sCV[0][0], wave, lane);
  __syncthreads();

  // Phase 3a: skip-gram deno loss (pos + 10 neg dots per row), wave-per-dot, b128 loads
  float denoAcc = 0.f;
  for (int d = wave; d < ROWS * (KNEG + 1); d += WAVES) {
    int r = d / (KNEG + 1);
    int j = d - r * (KNEG + 1);
    int gid = (j == 0) ? ctx_ids[row0 + r]
                       : neg_ids[(row0 + r) * KNEG + (j - 1)];
    const float4* __restrict__ crow4 =
        (const float4*)(context_emb + (size_t)gid * EMBED);
    const float4* __restrict__ cv4 = (const float4*)&sCV[r][0];
    float p = 0.f;
    for (int c4 = lane; c4 < 75; c4 += 32) {
      float4 x = cv4[c4];
      float4 y = crow4[c4];
      p += x.x * y.x + x.y * y.y + x.z * y.z + x.w * y.w;
    }
    p = waveRedSum(p);
    if (lane == 0) {
      float s = fminf(fmaxf(p, -10.f), 10.f);
      // -logsigmoid(s) = log1p(exp(-s)); -logsigmoid(-s) = log1p(exp(s))
      denoAcc += (j == 0) ? log1pf(expf(-s)) : log1pf(expf(s));
    }
  }

  // Phase 3b: classifier cono loss (2 logits per row), wave-per-row, b128 loads
  float conoAcc = 0.f;
  for (int r = wave; r < ROWS; r += WAVES) {
    const float4* __restrict__ e4 = (const float4*)&sEnc[r][0];
    const float4* __restrict__ w04 = (const float4*)W_cls;
    const float4* __restrict__ w14 = (const float4*)(W_cls + HIDDEN);
    float p0 = 0.f, p1 = 0.f;
    for (int c4 = lane; c4 < 75; c4 += 32) {
      float4 e = e4[c4];
      float4 a = w04[c4];
      float4 b = w14[c4];
      p0 += e.x * a.x + e.y * a.y + e.z * a.z + e.w * a.w;
      p1 += e.x * b.x + e.y * b.y + e.z * b.z + e.w * b.w;
    }
    p0 = waveRedSum(p0);
    p1 = waveRedSum(p1);
    if (lane == 0) {
      float l0 = p0 + b_cls[0], l1 = p1 + b_cls[1];
      float mx = fmaxf(l0, l1);
      float lse = mx + logf(expf(l0 - mx) + expf(l1 - mx));
      conoAcc += lse - ((labels[row0 + r] == 0) ? l0 : l1);
    }
  }

  if (lane == 0) { sD[wave] = denoAcc; sC[wave] = conoAcc; }
  __syncthreads();
  if (tid == 0) {
    float dsum = 0.f, csum = 0.f;
#pragma unroll
    for (int w = 0; w < WAVES; ++w) { dsum += sD[w]; csum += sC[w]; }
    part[blockIdx.x * 2 + 0] = dsum;
    part[blockIdx.x * 2 + 1] = csum;
  }
}

__global__ __launch_bounds__(256) void finalize(
    const float* __restrict__ part, float* __restrict__ out) {
  __shared__ float sd[256], sc[256];
  int t = threadIdx.x;
  float d = 0.f, c = 0.f;
  for (int i = t; i < NBLOCKS; i += 256) {
    d += part[2 * i];
    c += part[2 * i + 1];
  }
  sd[t] = d; sc[t] = c;
  __syncthreads();
  for (int s = 128; s > 0; s >>= 1) {
    if (t < s) { sd[t] += sd[t + s]; sc[t] += sc[t + s]; }
    __syncthreads();
  }
  if (t == 0) {
    float deno = fminf(fmaxf(sd[0] / (float)BATCHSZ, EPSV), 10.f);
    float cono = fminf(fmaxf(sc[0] / (float)BATCHSZ, EPSV), 10.f);
    out[0] = fmaxf(deno + cono, EPSV);
    out[1] = deno;
    out[2] = cono;
  }
}

extern "C" void kernel_launch(void* const* d_in, const int* in_sizes, int n_in,
                              void* d_out, int out_size, void* d_ws, size_t ws_size,
                              hipStream_t stream) {
  const int* center_ids = (const int*)d_in[0];
  const int* ctx_ids = (const int*)d_in[1];
  const int* neg_ids = (const int*)d_in[2];
  const int* labels = (const int*)d_in[3];
  const float* center_emb = (const float*)d_in[4];
  const float* context_emb = (const float*)d_in[5];
  const float* W_enc = (const float*)d_in[6];
  const float* b_enc = (const float*)d_in[7];
  const float* W_dec = (const float*)d_in[8];
  const float* b_dec = (const float*)d_in[9];
  const float* W_cls = (const float*)d_in[10];
  const float* b_cls = (const float*)d_in[11];
  float* part = (float*)d_ws;  // NBLOCKS * 2 floats

  fused_fwd<<<NBLOCKS, TPB, 0, stream>>>(
      center_ids, ctx_ids, neg_ids, labels, center_emb, context_emb,
      W_enc, b_enc, W_dec, b_dec, W_cls, b_cls, part);
  finalize<<<1, 256, 0, stream>>>(part, (float*)d_out);
}